// GraphSAGE_30116310680318
// MI455X (gfx1250) — compile-verified
//
#include <hip/hip_runtime.h>

// GraphSAGE forward for MI455X (gfx1250, wave32).
//   h   = relu(mean(x) @ W1_l + b1 + x @ W1_r)
//   out =      mean(h) @ W2_l + b2 + h @ W2_r
// Scatter phase: coalesced float4 row gathers + global_atomic_add_f32 (L2-resident).
// GEMM phase: V_WMMA_F32_16X16X4_F32, one wave per 16x16 output tile.

typedef __attribute__((ext_vector_type(2))) float v2f;
typedef __attribute__((ext_vector_type(8))) float v8f;

#define N_NODES 50000
#define FDIM    128

// ---------------------------------------------------------------- utilities
__global__ void zero_f32_kernel(float* __restrict__ p, int n) {
    int i = blockIdx.x * blockDim.x + threadIdx.x;
    if (i < n) p[i] = 0.0f;
}

__global__ void degree_kernel(const int* __restrict__ dst,
                              float* __restrict__ deg, int E) {
    int e = blockIdx.x * blockDim.x + threadIdx.x;
    if (e < E) unsafeAtomicAdd(&deg[dst[e]], 1.0f);
}

// One wave per edge: 32 lanes x float4 = one full 512B feature row.
__global__ void scatter_add_kernel(const float* __restrict__ x,
                                   const int* __restrict__ src,
                                   const int* __restrict__ dst,
                                   float* __restrict__ agg, int E) {
    int lane = threadIdx.x & 31;
    int e = blockIdx.x * (blockDim.x >> 5) + (threadIdx.x >> 5);
    if (e >= E) return;
    int s = src[e];
    int d = dst[e];
    const float4* row = reinterpret_cast<const float4*>(x + (long)s * FDIM);
    float4 v = row[lane];
    float* out = agg + (long)d * FDIM + lane * 4;
    unsafeAtomicAdd(out + 0, v.x);
    unsafeAtomicAdd(out + 1, v.y);
    unsafeAtomicAdd(out + 2, v.z);
    unsafeAtomicAdd(out + 3, v.w);
}

__global__ void mean_kernel(float* __restrict__ agg,
                            const float* __restrict__ deg, int n) {
    int i = blockIdx.x * blockDim.x + threadIdx.x;
    if (i < n) {
        int node = i >> 7;  // FDIM == 128
        agg[i] = agg[i] / fmaxf(deg[node], 1.0f);
    }
}

// ----------------------------------------------------- fused SAGE GEMM (WMMA)
// out[M x NC] = Xmean @ Wl + Xself @ Wr + bias   (optionally ReLU)
// Xmean/Xself are [M x K] (K==128), Wl/Wr are [K x NC].
// blockDim = (32, colTiles); gridDim.x = M/16 (M is a multiple of 16).
// A layout (16x4 f32): lane L -> row (L&15), VGPR0 = K + 2*(L>>4), VGPR1 = +1.
// B layout (4x16 f32): VGPR0 = row K + 2*(L>>4), col (L&15); VGPR1 = +1.
// C/D layout: VGPR i -> row i + 8*(L>>4), col (L&15).
__global__ void sage_gemm_wmma(const float* __restrict__ Xmean,
                               const float* __restrict__ Xself,
                               const float* __restrict__ Wl,
                               const float* __restrict__ Wr,
                               const float* __restrict__ bias,
                               float* __restrict__ out,
                               int M, int K, int NC, int do_relu) {
    const int lane = threadIdx.x;          // 0..31
    const int half = lane >> 4;            // 0 or 1
    const int l15  = lane & 15;
    const int rt   = blockIdx.x;           // row tile
    const int ct   = threadIdx.y;          // col tile

    const int rowA = rt * 16 + l15;        // M multiple of 16 -> always valid
    const int koff = half * 2;

    const int nGlobal = ct * 16 + l15;
    const float nMask = (nGlobal < NC) ? 1.0f : 0.0f;  // branchless col guard
    const int  nIdx   = (nGlobal < NC) ? nGlobal : 0;  // clamped (valid load)

    const float* xm = Xmean + (long)rowA * K;
    const float* xs = Xself + (long)rowA * K;

    v8f acc = {};
    for (int k = 0; k < K; k += 4) {
        const int kk = k + koff;
        v2f a, b;

        // mean @ Wl
        a.x = xm[kk];
        a.y = xm[kk + 1];
        b.x = Wl[(long)kk * NC + nIdx] * nMask;
        b.y = Wl[(long)(kk + 1) * NC + nIdx] * nMask;
        acc = __builtin_amdgcn_wmma_f32_16x16x4_f32(
                  false, a, false, b, (short)0, acc, false, false);

        // x @ Wr
        a.x = xs[kk];
        a.y = xs[kk + 1];
        b.x = Wr[(long)kk * NC + nIdx] * nMask;
        b.y = Wr[(long)(kk + 1) * NC + nIdx] * nMask;
        acc = __builtin_amdgcn_wmma_f32_16x16x4_f32(
                  false, a, false, b, (short)0, acc, false, false);
    }

    const float bv = bias[nIdx];
    if (nGlobal < NC) {
        #pragma unroll
        for (int i = 0; i < 8; ++i) {
            const int row = rt * 16 + i + half * 8;
            float v = acc[i] + bv;
            if (do_relu) v = fmaxf(v, 0.0f);
            out[(long)row * NC + nGlobal] = v;
        }
    }
}

// ------------------------------------------------------------------- launch
extern "C" void kernel_launch(void* const* d_in, const int* in_sizes, int n_in,
                              void* d_out, int out_size, void* d_ws, size_t ws_size,
                              hipStream_t stream) {
    const float* x   = (const float*)d_in[0];   // [N, 128]
    const int*   ei  = (const int*)  d_in[1];   // [2, E]
    const float* W1l = (const float*)d_in[2];   // [128, 128]
    const float* b1  = (const float*)d_in[3];   // [128]
    const float* W1r = (const float*)d_in[4];   // [128, 128]
    const float* W2l = (const float*)d_in[5];   // [128, 40]
    const float* b2  = (const float*)d_in[6];   // [40]
    const float* W2r = (const float*)d_in[7];   // [128, 40]
    float* out = (float*)d_out;                 // [N, 40]

    const int E = in_sizes[1] / 2;
    const int* src = ei;
    const int* dst = ei + E;

    // workspace layout (tightly packed, 256B-aligned chunks)
    char* ws = (char*)d_ws;
    size_t off = 0;
    float* deg = (float*)(ws + off); off += ((size_t)N_NODES * 4 + 255) & ~255ull;
    float* agg = (float*)(ws + off); off += ((size_t)N_NODES * FDIM * 4 + 255) & ~255ull;
    float* h   = (float*)(ws + off);

    const int NF = N_NODES * FDIM;  // 6.4M

    // ---- degree (shared by both layers)
    zero_f32_kernel<<<(N_NODES + 255) / 256, 256, 0, stream>>>(deg, N_NODES);
    degree_kernel<<<(E + 255) / 256, 256, 0, stream>>>(dst, deg, E);

    // ---- layer 1
    zero_f32_kernel<<<(NF + 255) / 256, 256, 0, stream>>>(agg, NF);
    scatter_add_kernel<<<(E + 7) / 8, 256, 0, stream>>>(x, src, dst, agg, E);
    mean_kernel<<<(NF + 255) / 256, 256, 0, stream>>>(agg, deg, NF);
    {
        dim3 blk(32, 8);                    // 8 col tiles (NC = 128)
        dim3 grd(N_NODES / 16);             // 3125 row tiles
        sage_gemm_wmma<<<grd, blk, 0, stream>>>(agg, x, W1l, W1r, b1, h,
                                                N_NODES, FDIM, FDIM, /*relu=*/1);
    }

    // ---- layer 2
    zero_f32_kernel<<<(NF + 255) / 256, 256, 0, stream>>>(agg, NF);
    scatter_add_kernel<<<(E + 7) / 8, 256, 0, stream>>>(h, src, dst, agg, E);
    mean_kernel<<<(NF + 255) / 256, 256, 0, stream>>>(agg, deg, NF);
    {
        dim3 blk(32, 3);                    // 3 col tiles cover NC = 40
        dim3 grd(N_NODES / 16);
        sage_gemm_wmma<<<grd, blk, 0, stream>>>(agg, h, W2l, W2r, b2, out,
                                                N_NODES, FDIM, 40, /*relu=*/0);
    }
}